// CropPrompter_395136991516
// MI455X (gfx1250) — compile-verified
//
#include <hip/hip_runtime.h>
#include <stdint.h>

#ifndef __has_builtin
#define __has_builtin(x) 0
#endif

#define NB 8
#define NC 3
#define NT 16
#define NH 512
#define NW 512
#define NS 224
#define RESIZE_MAX_F 1024.0f
#define ROWS_PER_TILE 16
#define NTILES (NS / ROWS_PER_TILE)   /* 14 */
#define MAXR 17                       /* max source rows per 16-output-row tile (scale<=1) */
#define MAXC 228                      /* max aligned source cols for 224 outputs */
#define THREADS 256
#define VEC4_PER_ROW (NS / 4)                       /* 56 */
#define VEC4_PER_TILE (ROWS_PER_TILE * VEC4_PER_ROW) /* 896 */

typedef unsigned int u32x4 __attribute__((ext_vector_type(4)));
typedef int          i32x4 __attribute__((ext_vector_type(4)));
typedef int          i32x8 __attribute__((ext_vector_type(8)));
typedef float        f32x4 __attribute__((ext_vector_type(4)));

// Tensor Data Mover path (confirmed compiling in round 1: TDM builtin lowered,
// ds histogram shows zero LDS stores -> DMA is the only writer of the tile).
// Arity hedge: ROCm7.2 = 5 args; amdgpu-toolchain (ships the TDM header) = 6.
#if defined(__AMDGCN__) && __has_builtin(__builtin_amdgcn_tensor_load_to_lds) && \
    __has_builtin(__builtin_amdgcn_s_wait_tensorcnt)
#define USE_TDM 1
#if defined(__has_include)
#if __has_include(<hip/amd_detail/amd_gfx1250_TDM.h>)
#define TDM_SIX_ARG 1
#endif
#endif
#else
#define USE_TDM 0
#endif

__device__ __forceinline__ int rfl(int v) {
#if defined(__AMDGCN__)
  return __builtin_amdgcn_readfirstlane(v);
#else
  return v;
#endif
}

__global__ __launch_bounds__(THREADS) void crop_prompt_kernel(
    const float* __restrict__ x, const int* __restrict__ cam_views,
    const float* __restrict__ resize, const float* __restrict__ y_offset,
    const float* __restrict__ x_offset, float* __restrict__ out) {
  __shared__ float tile[MAXR * MAXC];

  const int tid = threadIdx.x;
  const int blk = blockIdx.x;
  const int tixr = blk % NTILES;  // which 16-row tile of the 224-row crop
  const int f    = blk / NTILES;  // frame index: (b*NC + c)*NT + t
  const int b    = f / (NC * NT);

  // Per-clip learned crop params (uniform across the block -> scalar loads).
  const int v   = cam_views[b];
  const float r = floorf(fminf(fmaxf(resize[v], (float)NH), RESIZE_MAX_F));
  const float yo = floorf(fminf(fmaxf(y_offset[v], 0.0f), r - (float)NS));
  const float xo = floorf(fminf(fmaxf(x_offset[v], 0.0f), r - (float)NS));
  const float scale = (float)NH / r;  // in [0.5, 1]

  const int s0 = tixr * ROWS_PER_TILE;

  // src = max((off + i + 0.5)*scale - 0.5, 0); i0 = clamp(floor(src), 0, 511)
  auto coord0 = [&](float off, int i) -> int {
    float s = fmaxf((off + (float)i + 0.5f) * scale - 0.5f, 0.0f);
    int i0 = (int)s;            // trunc == floor for s >= 0
    return (i0 > NH - 1) ? (NH - 1) : i0;
  };

  // Source window for this tile (block-uniform).
  const int rowLo = coord0(yo, s0);
  int rowHi = coord0(yo, s0 + ROWS_PER_TILE - 1) + 1;
  if (rowHi > NH - 1) rowHi = NH - 1;
  const int nRows = rowHi - rowLo + 1;               // <= 17

  const int colLo = coord0(xo, 0) & ~3;              // 16B align for clean DMA
  int colHi = coord0(xo, NS - 1) + 1;
  if (colHi > NW - 1) colHi = NW - 1;
  const int nCols = ((colHi - colLo + 1) + 3) & ~3;  // <= 228, colLo+nCols <= 512

  const float* src = x + (size_t)f * ((size_t)NH * NW);

#if USE_TDM
  // One wave issues a single TDM 2D-tile DMA: nRows x nCols f32, row stride 512,
  // packed contiguously into LDS (stride nCols). Tracked with TENSORcnt.
  if (tid < 32) {
    const unsigned long long ga =
        (unsigned long long)(uintptr_t)(src + (size_t)rowLo * NW + colLo);
    const unsigned ldsa = (unsigned)(uintptr_t)(&tile[0]);  // LDS byte offset
    const unsigned td0 = (unsigned)nCols, td1 = (unsigned)nRows;

    // D# group 0 (ISA 8.3): count=1 | lds_addr | global_addr[56:0] | type=2
    u32x4 g0;
    g0.x = (unsigned)rfl(1);
    g0.y = (unsigned)rfl((int)ldsa);
    g0.z = (unsigned)rfl((int)(unsigned)ga);
    g0.w = (unsigned)rfl((int)((unsigned)(ga >> 32) | (2u << 30)));

    // D# group 1 (ISA 8.4): data_size=4B; tensor_dim0/1 = tile dims (tile is
    // in-bounds by construction, so no OOB zero-fill); tile_dim0=nCols,
    // tile_dim1=nRows, tile_dim2=0; tensor_dim0_stride=512 elements.
    i32x8 g1;
    g1[0] = rfl((int)(2u << 16));
    g1[1] = rfl((int)((td0 & 0xFFFFu) << 16));
    g1[2] = rfl((int)((td0 >> 16) | ((td1 & 0xFFFFu) << 16)));
    g1[3] = rfl((int)((td1 >> 16) | (td0 << 16)));
    g1[4] = rfl((int)td1);
    g1[5] = rfl((int)NW);
    g1[6] = 0;
    g1[7] = 0;

    i32x4 g2 = {0, 0, 0, 0};  // tensor_dim2=0, tile_dim3=0 (2D tensor)
    i32x4 g3 = {0, 0, 0, 0};

#if defined(TDM_SIX_ARG)
    i32x8 g4 = {0, 0, 0, 0, 0, 0, 0, 0};
    __builtin_amdgcn_tensor_load_to_lds(g0, g1, g2, g3, g4, 0);
#else
    __builtin_amdgcn_tensor_load_to_lds(g0, g1, g2, g3, 0);
#endif
    __builtin_amdgcn_s_wait_tensorcnt(0);
  }
  __syncthreads();
  {
    // The DMA is the only writer of `tile`: keep LLVM from folding the loads.
    void* tp = (void*)tile;
    asm volatile("" : "+v"(tp) : : "memory");
  }
#else
  // Fallback: cooperative synchronous stage (global -> LDS), same layout.
  for (int rr = 0; rr < nRows; ++rr) {
    const float* rsrc = src + (size_t)(rowLo + rr) * NW + colLo;
    float* rdst = tile + rr * nCols;
    for (int cc = tid; cc < nCols; cc += THREADS) rdst[cc] = rsrc[cc];
  }
  __syncthreads();
#endif

  // Each thread produces float4's of 4 consecutive output pixels (same row):
  // y-coordinate math amortized 4x, output written with 16B-aligned b128
  // stores (896 vec4 per tile; contiguous, coalesced).
  const size_t obase = (size_t)f * ((size_t)NS * NS);
  for (int p = tid; p < VEC4_PER_TILE; p += THREADS) {
    const int srow = p / VEC4_PER_ROW;            // 0..15
    const int j0 = (p - srow * VEC4_PER_ROW) * 4; // first of 4 columns
    const int sg = s0 + srow;

    // Vertical coords: shared by the 4 pixels.
    float sy = fmaxf((yo + (float)sg + 0.5f) * scale - 0.5f, 0.0f);
    int y0 = (int)sy; if (y0 > NH - 1) y0 = NH - 1;
    const float wy = sy - (float)y0;
    int y1 = y0 + 1; if (y1 > NH - 1) y1 = NH - 1;
    const int ry0 = (y0 - rowLo) * nCols;
    const int ry1 = (y1 - rowLo) * nCols;
    const float omwy = 1.0f - wy;

    f32x4 res;
#pragma unroll
    for (int e = 0; e < 4; ++e) {
      const int j = j0 + e;
      float sx = fmaxf((xo + (float)j + 0.5f) * scale - 0.5f, 0.0f);
      int x0 = (int)sx; if (x0 > NW - 1) x0 = NW - 1;
      const float wx = sx - (float)x0;
      int x1 = x0 + 1; if (x1 > NW - 1) x1 = NW - 1;
      const int cx0 = x0 - colLo;
      const int cx1 = x1 - colLo;

      const float a  = tile[ry0 + cx0];
      const float bq = tile[ry0 + cx1];
      const float cq = tile[ry1 + cx0];
      const float d  = tile[ry1 + cx1];

      // rows = clip[y0]*(1-wy) + clip[y1]*wy ; out = rows[x0]*(1-wx)+rows[x1]*wx
      const float rowA = a * omwy + cq * wy;
      const float rowB = bq * omwy + d * wy;
      res[e] = rowA * (1.0f - wx) + rowB * wx;
    }
    *(f32x4*)(out + obase + (size_t)sg * NS + j0) = res;  // global_store_b128
  }
}

extern "C" void kernel_launch(void* const* d_in, const int* in_sizes, int n_in,
                              void* d_out, int out_size, void* d_ws, size_t ws_size,
                              hipStream_t stream) {
  const float* x        = (const float*)d_in[0];
  const int*   cam      = (const int*)d_in[1];
  const float* resize   = (const float*)d_in[2];
  const float* y_offset = (const float*)d_in[3];
  const float* x_offset = (const float*)d_in[4];
  float* out = (float*)d_out;

  const int blocks = NB * NC * NT * NTILES;  // 5376
  crop_prompt_kernel<<<blocks, THREADS, 0, stream>>>(x, cam, resize, y_offset,
                                                     x_offset, out);
}